// ALEncoder_84250078478931
// MI455X (gfx1250) — compile-verified
//
#include <hip/hip_runtime.h>
#include <hip/hip_bf16.h>
#include <math.h>
#include <stdint.h>

#define DEV __device__ __forceinline__

typedef float v2f __attribute__((ext_vector_type(2)));
typedef float v8f __attribute__((ext_vector_type(8)));

constexpr int N_ACTOR_C = 16384;
constexpr int KNB = 32;        // lane neighbors per actor
constexpr int D = 64;          // feature dim
constexpr int BA = 16;         // actors per block (one WMMA M tile)
constexpr int NT = 128;        // 4 wave32s per block
constexpr int LS = 68;         // padded LDS row stride (floats) for staging tiles
constexpr float EPSV = 1e-5f;

// ---------------- WMMA helpers (fp32, V_WMMA_F32_16X16X4_F32 chained K=64) ----
// A: 16x64 row-major (stride lda) in LDS; B: 64x16 slab of row-major 64x64
// weight (stride ldb) in LDS; returns 16x16 f32 accumulator.
DEV v8f wmma_tile_k64(const float* A, int lda, const float* B, int ldb, v8f acc) {
  const int lane = threadIdx.x & 31;
  const int hi = lane >> 4;   // 0: K pair {0,1}; 1: K pair {2,3} of each 4-slice
  const int l16 = lane & 15;
#pragma unroll
  for (int kk = 0; kk < 16; ++kk) {
    const int k0 = kk * 4 + hi * 2;
    v2f a, b;
    a.x = A[l16 * lda + k0];
    a.y = A[l16 * lda + k0 + 1];
    b.x = B[(k0) * ldb + l16];
    b.y = B[(k0 + 1) * ldb + l16];
    acc = __builtin_amdgcn_wmma_f32_16x16x4_f32(false, a, false, b, (short)0,
                                                acc, false, false);
  }
  return acc;
}

// C/D layout: VGPR i -> row (i + 8*hi), col l16
DEV void store_tile(float* dst, int ldd, v8f c) {
  const int lane = threadIdx.x & 31;
  const int hi = lane >> 4, l16 = lane & 15;
#pragma unroll
  for (int i = 0; i < 8; ++i) dst[(i + 8 * hi) * ldd + l16] = c[i];
}

DEV v8f vzero() {
  v8f z = {0.f, 0.f, 0.f, 0.f, 0.f, 0.f, 0.f, 0.f};
  return z;
}

// ---------------- CDNA5 async global->LDS copy (ASYNCcnt path) --------------
DEV void async_copy(const float* g, float* l, int n) {
  for (int i = threadIdx.x * 4; i < n; i += NT * 4) {
    unsigned lds = (unsigned)(size_t)(l + i);
    const float* ga = g + i;
    asm volatile("global_load_async_to_lds_b128 %0, %1, off"
                 :
                 : "v"(lds), "v"(ga)
                 : "memory");
  }
}

DEV void wait_async() { asm volatile("s_wait_asynccnt 0" ::: "memory"); }

DEV void cp(float* d, const float* s, int n) {
  for (int i = threadIdx.x; i < n; i += NT) d[i] = s[i];
}

DEV float dot8(const float* a, const float* b) {
  float s = 0.f;
#pragma unroll
  for (int i = 0; i < 8; ++i) s += a[i] * b[i];
  return s;
}

DEV float sigmoidf_(float x) { return 1.f / (1.f + __expf(-x)); }

struct P {
  const float *x_lane, *x_actor, *edge_attr;
  const float *le_n_W1, *le_n_b1, *le_n_g1, *le_n_be1, *le_n_W2, *le_n_b2;
  const float *le_e_W1, *le_e_b1, *le_e_g1, *le_e_be1, *le_e_W2, *le_e_b2;
  const float *ag_g1, *ag_b1, *ag_W, *ag_b, *ag_g2, *ag_b2;
  const float *emb_is, *emb_turn, *emb_tc;
  const float *Wq, *Wk, *Wv, *Wo, *W_self, *W_ih, *W_hh;
  const float *b_self, *b_ih, *b_hh;
  const float *n1_g, *n1_b, *n2_g, *n2_b;
  const float *mlp_W1, *mlp_b1, *mlp_W2, *mlp_b2, *mlp_W3, *mlp_b3;
  const int *nbr_idx, *is_inter, *turn_dir, *traffic;
  float* out;
};

__global__ __launch_bounds__(NT) void alenc_kernel(P p) {
  extern __shared__ float smem[];
  // ---- LDS layout ----
  float* sWB = smem;                     // 7*4096 big-weight region (phased)
  float* sp = smem + 7 * 4096;
  float* sW1n = sp; sp += 128; float* sB1n = sp; sp += 64;
  float* sG1n = sp; sp += 64;  float* sBe1n = sp; sp += 64;
  float* sB2n = sp; sp += 64;
  float* sW1e = sp; sp += 128; float* sB1e = sp; sp += 64;
  float* sG1e = sp; sp += 64;  float* sBe1e = sp; sp += 64;
  float* sB2e = sp; sp += 64;
  float* sAgG1 = sp; sp += 64; float* sAgB1 = sp; sp += 64;
  float* sAgB = sp; sp += 64;  float* sAgG2 = sp; sp += 64;
  float* sAgB2 = sp; sp += 64;
  float* sEmbIs = sp; sp += 128; float* sEmbTurn = sp; sp += 192;
  float* sEmbTc = sp; sp += 128;
  float* sBself = sp; sp += 64; float* sBih = sp; sp += 64;
  float* sBhh = sp; sp += 64;
  float* sN1g = sp; sp += 64; float* sN1b = sp; sp += 64;
  float* sN2g = sp; sp += 64; float* sN2b = sp; sp += 64;
  float* sMb1 = sp; sp += 64; float* sMb2 = sp; sp += 64;
  float* sMb3 = sp; sp += 64;
  float* sXa = sp;     sp += BA * LS;
  float* sCenter = sp; sp += BA * LS;
  float* sQ = sp;      sp += BA * LS;
  float* sAttn = sp;   sp += BA * LS;
  float* sK = sp;      sp += KNB * LS;
  float* sV = sp;      sp += KNB * LS;
  float* sSt1 = sp;    sp += BA * LS;
  float* sSt2 = sp;    sp += BA * LS;
  float* sSt3 = sp;    sp += BA * LS;

  __shared__ int metaII[16], metaTT[16], metaCC[16];

  const int tid = threadIdx.x;
  const int wave = tid >> 5;
  const int n0 = wave * 16;              // this wave's output-column slab
  const int ab = blockIdx.x * BA;

  // ---- Phase A: stage edge-phase weights via async tensor path ----
  async_copy(p.le_n_W2, sWB + 0 * 4096, 4096);
  async_copy(p.le_e_W2, sWB + 1 * 4096, 4096);
  async_copy(p.ag_W,    sWB + 2 * 4096, 4096);
  async_copy(p.Wq,      sWB + 3 * 4096, 4096);
  async_copy(p.Wk,      sWB + 4 * 4096, 4096);
  async_copy(p.Wv,      sWB + 5 * 4096, 4096);
  cp(sW1n, p.le_n_W1, 128); cp(sB1n, p.le_n_b1, 64);
  cp(sG1n, p.le_n_g1, 64);  cp(sBe1n, p.le_n_be1, 64);
  cp(sB2n, p.le_n_b2, 64);
  cp(sW1e, p.le_e_W1, 128); cp(sB1e, p.le_e_b1, 64);
  cp(sG1e, p.le_e_g1, 64);  cp(sBe1e, p.le_e_be1, 64);
  cp(sB2e, p.le_e_b2, 64);
  cp(sAgG1, p.ag_g1, 64); cp(sAgB1, p.ag_b1, 64); cp(sAgB, p.ag_b, 64);
  cp(sAgG2, p.ag_g2, 64); cp(sAgB2, p.ag_b2, 64);
  cp(sEmbIs, p.emb_is, 128); cp(sEmbTurn, p.emb_turn, 192);
  cp(sEmbTc, p.emb_tc, 128);
  cp(sBself, p.b_self, 64); cp(sBih, p.b_ih, 64); cp(sBhh, p.b_hh, 64);
  cp(sN1g, p.n1_g, 64); cp(sN1b, p.n1_b, 64);
  cp(sN2g, p.n2_g, 64); cp(sN2b, p.n2_b, 64);
  cp(sMb1, p.mlp_b1, 64); cp(sMb2, p.mlp_b2, 64); cp(sMb3, p.mlp_b3, 64);
  wait_async();
  __syncthreads();

  // ---- Prologue: center = LN(x_actor, n1); q = center @ Wq ----
  for (int i = tid; i < BA * D; i += NT) {
    int r = i >> 6, d = i & 63;
    sXa[r * LS + d] = p.x_actor[(size_t)(ab + r) * D + d];
  }
  __syncthreads();
  {
    const int r = tid >> 3, sub = tid & 7;
    float v[8], s = 0.f, sq = 0.f;
#pragma unroll
    for (int j = 0; j < 8; ++j) {
      v[j] = sXa[r * LS + sub * 8 + j];
      s += v[j]; sq += v[j] * v[j];
    }
#pragma unroll
    for (int m = 1; m < 8; m <<= 1) {
      s += __shfl_xor(s, m, 32);
      sq += __shfl_xor(sq, m, 32);
    }
    const float mu = s * (1.f / 64.f);
    const float inv = rsqrtf(sq * (1.f / 64.f) - mu * mu + EPSV);
#pragma unroll
    for (int j = 0; j < 8; ++j) {
      int d = sub * 8 + j;
      sCenter[r * LS + d] = (v[j] - mu) * inv * sN1g[d] + sN1b[d];
    }
  }
  __syncthreads();
  {
    v8f acc = vzero();
    acc = wmma_tile_k64(sCenter, LS, sWB + 3 * 4096 + n0, D, acc);
    store_tile(sQ + n0, LS, acc);
  }
  __syncthreads();

  // ---- Edge phase: per actor, 2 x 16-row WMMA tiles, then attention -------
  for (int a = 0; a < BA; ++a) {
    const int ga = ab + a;
    for (int hf = 0; hf < 2; ++hf) {
      const int r = tid >> 3, sub = tid & 7;
      const int e = ga * KNB + hf * 16 + r;
      const int nj = p.nbr_idx[e];
      const float xj0 = p.x_lane[2 * nj], xj1 = p.x_lane[2 * nj + 1];
      const float e0 = p.edge_attr[(size_t)e * 2], e1 = p.edge_attr[(size_t)e * 2 + 1];
      if (sub == 0) {
        metaII[r] = p.is_inter[nj];
        metaTT[r] = p.turn_dir[nj];
        metaCC[r] = p.traffic[nj];
      }
      // node & edge branch layer-1 (tiny 2->64 GEMV in VALU) + LN + relu
      float vn[8], ve[8];
      float sn = 0.f, sqn = 0.f, se = 0.f, sqe = 0.f;
#pragma unroll
      for (int j = 0; j < 8; ++j) {
        int d = sub * 8 + j;
        vn[j] = xj0 * sW1n[d] + xj1 * sW1n[64 + d] + sB1n[d];
        ve[j] = e0 * sW1e[d] + e1 * sW1e[64 + d] + sB1e[d];
        sn += vn[j]; sqn += vn[j] * vn[j];
        se += ve[j]; sqe += ve[j] * ve[j];
      }
#pragma unroll
      for (int m = 1; m < 8; m <<= 1) {
        sn += __shfl_xor(sn, m, 32); sqn += __shfl_xor(sqn, m, 32);
        se += __shfl_xor(se, m, 32); sqe += __shfl_xor(sqe, m, 32);
      }
      const float mun = sn * (1.f / 64.f);
      const float invn = rsqrtf(sqn * (1.f / 64.f) - mun * mun + EPSV);
      const float mue = se * (1.f / 64.f);
      const float inve = rsqrtf(sqe * (1.f / 64.f) - mue * mue + EPSV);
#pragma unroll
      for (int j = 0; j < 8; ++j) {
        int d = sub * 8 + j;
        sSt1[r * LS + d] = fmaxf((vn[j] - mun) * invn * sG1n[d] + sBe1n[d], 0.f);
        sSt2[r * LS + d] = fmaxf((ve[j] - mue) * inve * sG1e[d] + sBe1e[d], 0.f);
      }
      __syncthreads();
      // s = hn@W2n + he@W2e  (WMMA, per-wave 16-col slab)
      {
        v8f acc = vzero();
        acc = wmma_tile_k64(sSt1, LS, sWB + 0 * 4096 + n0, D, acc);
        acc = wmma_tile_k64(sSt2, LS, sWB + 1 * 4096 + n0, D, acc);
        store_tile(sSt3 + n0, LS, acc);
      }
      __syncthreads();
      // s += biases + embeddings; t = relu(LN(s, ag1))
      {
        float v[8], s = 0.f, sq = 0.f;
        const int ii = metaII[r], tt = metaTT[r], cc = metaCC[r];
#pragma unroll
        for (int j = 0; j < 8; ++j) {
          int d = sub * 8 + j;
          v[j] = sSt3[r * LS + d] + sB2n[d] + sB2e[d] + sEmbIs[ii * 64 + d] +
                 sEmbTurn[tt * 64 + d] + sEmbTc[cc * 64 + d];
          s += v[j]; sq += v[j] * v[j];
        }
#pragma unroll
        for (int m = 1; m < 8; m <<= 1) {
          s += __shfl_xor(s, m, 32); sq += __shfl_xor(sq, m, 32);
        }
        const float mu = s * (1.f / 64.f);
        const float inv = rsqrtf(sq * (1.f / 64.f) - mu * mu + EPSV);
#pragma unroll
        for (int j = 0; j < 8; ++j) {
          int d = sub * 8 + j;
          sSt1[r * LS + d] = fmaxf((v[j] - mu) * inv * sAgG1[d] + sAgB1[d], 0.f);
        }
      }
      __syncthreads();
      // u = t @ ag_W
      {
        v8f acc = vzero();
        acc = wmma_tile_k64(sSt1, LS, sWB + 2 * 4096 + n0, D, acc);
        store_tile(sSt2 + n0, LS, acc);
      }
      __syncthreads();
      // lane = LN(u + ag_b, ag2)
      {
        float v[8], s = 0.f, sq = 0.f;
#pragma unroll
        for (int j = 0; j < 8; ++j) {
          int d = sub * 8 + j;
          v[j] = sSt2[r * LS + d] + sAgB[d];
          s += v[j]; sq += v[j] * v[j];
        }
#pragma unroll
        for (int m = 1; m < 8; m <<= 1) {
          s += __shfl_xor(s, m, 32); sq += __shfl_xor(sq, m, 32);
        }
        const float mu = s * (1.f / 64.f);
        const float inv = rsqrtf(sq * (1.f / 64.f) - mu * mu + EPSV);
#pragma unroll
        for (int j = 0; j < 8; ++j) {
          int d = sub * 8 + j;
          sSt3[r * LS + d] = (v[j] - mu) * inv * sAgG2[d] + sAgB2[d];
        }
      }
      __syncthreads();
      // k = lane@Wk ; v = lane@Wv  -> per-actor K/V tiles
      {
        v8f acc = vzero();
        acc = wmma_tile_k64(sSt3, LS, sWB + 4 * 4096 + n0, D, acc);
        store_tile(sK + hf * 16 * LS + n0, LS, acc);
        v8f acv = vzero();
        acv = wmma_tile_k64(sSt3, LS, sWB + 5 * 4096 + n0, D, acv);
        store_tile(sV + hf * 16 * LS + n0, LS, acv);
      }
      __syncthreads();
    }
    // ---- attention for actor a: 8 heads x 16 lanes, K=32 softmax ----
    {
      const int h = tid >> 4, r = tid & 15;
      const float* qp = sQ + a * LS + h * 8;
      const float sc = 0.3535533905932738f;  // 1/sqrt(8)
      float s0 = dot8(qp, sK + r * LS + h * 8) * sc;
      float s1 = dot8(qp, sK + (r + 16) * LS + h * 8) * sc;
      float mx = fmaxf(s0, s1);
#pragma unroll
      for (int m = 1; m < 16; m <<= 1) mx = fmaxf(mx, __shfl_xor(mx, m, 32));
      float p0 = __expf(s0 - mx), p1 = __expf(s1 - mx);
      float l = p0 + p1;
#pragma unroll
      for (int m = 1; m < 16; m <<= 1) l += __shfl_xor(l, m, 32);
      float av[8];
#pragma unroll
      for (int d = 0; d < 8; ++d)
        av[d] = p0 * sV[r * LS + h * 8 + d] + p1 * sV[(r + 16) * LS + h * 8 + d];
#pragma unroll
      for (int m = 1; m < 16; m <<= 1) {
#pragma unroll
        for (int d = 0; d < 8; ++d) av[d] += __shfl_xor(av[d], m, 32);
      }
      if (r == 0) {
#pragma unroll
        for (int d = 0; d < 8; ++d) sAttn[a * LS + h * 8 + d] = av[d] / l;
      }
    }
    __syncthreads();
  }

  // ---- Phase B: stage actor-phase weights (reuse region) ----
  async_copy(p.Wo,     sWB + 0 * 4096, 4096);
  async_copy(p.W_self, sWB + 1 * 4096, 4096);
  async_copy(p.W_ih,   sWB + 2 * 4096, 4096);
  async_copy(p.W_hh,   sWB + 3 * 4096, 4096);
  async_copy(p.mlp_W1, sWB + 4 * 4096, 4096);
  async_copy(p.mlp_W2, sWB + 5 * 4096, 4096);
  async_copy(p.mlp_W3, sWB + 6 * 4096, 4096);
  wait_async();
  __syncthreads();

  // ao = attn @ Wo
  {
    v8f acc = vzero();
    acc = wmma_tile_k64(sAttn, LS, sWB + 0 * 4096 + n0, D, acc);
    store_tile(sSt1 + n0, LS, acc);
  }
  __syncthreads();
  // gatepre = ao@W_ih + center@W_hh ; cs = center@W_self
  {
    v8f acc = vzero();
    acc = wmma_tile_k64(sSt1, LS, sWB + 2 * 4096 + n0, D, acc);
    acc = wmma_tile_k64(sCenter, LS, sWB + 3 * 4096 + n0, D, acc);
    store_tile(sSt2 + n0, LS, acc);
    v8f ac2 = vzero();
    ac2 = wmma_tile_k64(sCenter, LS, sWB + 1 * 4096 + n0, D, ac2);
    store_tile(sSt3 + n0, LS, ac2);
  }
  __syncthreads();
  // gate/msg; x1 = x_actor + msg; hN = LN(x1, n2)
  {
    const int r = tid >> 3, sub = tid & 7;
    float x1v[8], s = 0.f, sq = 0.f;
#pragma unroll
    for (int j = 0; j < 8; ++j) {
      int d = sub * 8 + j;
      float ao = sSt1[r * LS + d];
      float g = sigmoidf_(sSt2[r * LS + d] + sBih[d] + sBhh[d]);
      float cs = sSt3[r * LS + d] + sBself[d];
      float msg = ao + g * (cs - ao);
      x1v[j] = sXa[r * LS + d] + msg;
      s += x1v[j]; sq += x1v[j] * x1v[j];
    }
#pragma unroll
    for (int m = 1; m < 8; m <<= 1) {
      s += __shfl_xor(s, m, 32); sq += __shfl_xor(sq, m, 32);
    }
    const float mu = s * (1.f / 64.f);
    const float inv = rsqrtf(sq * (1.f / 64.f) - mu * mu + EPSV);
#pragma unroll
    for (int j = 0; j < 8; ++j) {
      int d = sub * 8 + j;
      sXa[r * LS + d] = x1v[j];                                  // x1
      sSt2[r * LS + d] = (x1v[j] - mu) * inv * sN2g[d] + sN2b[d]; // hN
    }
  }
  __syncthreads();
  // f1 = relu(hN@mlp_W1 + b1)
  {
    v8f acc = vzero();
    acc = wmma_tile_k64(sSt2, LS, sWB + 4 * 4096 + n0, D, acc);
    const int lane = tid & 31, hi = lane >> 4, l16 = lane & 15;
#pragma unroll
    for (int i = 0; i < 8; ++i)
      sSt3[(i + 8 * hi) * LS + n0 + l16] = fmaxf(acc[i] + sMb1[n0 + l16], 0.f);
  }
  __syncthreads();
  // f2 = relu(f1@mlp_W2 + b2)
  {
    v8f acc = vzero();
    acc = wmma_tile_k64(sSt3, LS, sWB + 5 * 4096 + n0, D, acc);
    const int lane = tid & 31, hi = lane >> 4, l16 = lane & 15;
#pragma unroll
    for (int i = 0; i < 8; ++i)
      sSt2[(i + 8 * hi) * LS + n0 + l16] = fmaxf(acc[i] + sMb2[n0 + l16], 0.f);
  }
  __syncthreads();
  // out = x1 + f2@mlp_W3 + b3
  {
    v8f acc = vzero();
    acc = wmma_tile_k64(sSt2, LS, sWB + 6 * 4096 + n0, D, acc);
    const int lane = tid & 31, hi = lane >> 4, l16 = lane & 15;
#pragma unroll
    for (int i = 0; i < 8; ++i) {
      const int row = i + 8 * hi, col = n0 + l16;
      p.out[(size_t)(ab + row) * D + col] =
          acc[i] + sMb3[col] + sXa[row * LS + col];
    }
  }
}

extern "C" void kernel_launch(void* const* d_in, const int* in_sizes, int n_in,
                              void* d_out, int out_size, void* d_ws, size_t ws_size,
                              hipStream_t stream) {
  (void)in_sizes; (void)n_in; (void)out_size; (void)d_ws; (void)ws_size;
  P p;
  p.x_lane   = (const float*)d_in[0];
  p.x_actor  = (const float*)d_in[1];
  p.edge_attr= (const float*)d_in[2];
  p.le_n_W1  = (const float*)d_in[3];
  p.le_n_b1  = (const float*)d_in[4];
  p.le_n_g1  = (const float*)d_in[5];
  p.le_n_be1 = (const float*)d_in[6];
  p.le_n_W2  = (const float*)d_in[7];
  p.le_n_b2  = (const float*)d_in[8];
  p.le_e_W1  = (const float*)d_in[9];
  p.le_e_b1  = (const float*)d_in[10];
  p.le_e_g1  = (const float*)d_in[11];
  p.le_e_be1 = (const float*)d_in[12];
  p.le_e_W2  = (const float*)d_in[13];
  p.le_e_b2  = (const float*)d_in[14];
  p.ag_g1    = (const float*)d_in[15];
  p.ag_b1    = (const float*)d_in[16];
  p.ag_W     = (const float*)d_in[17];
  p.ag_b     = (const float*)d_in[18];
  p.ag_g2    = (const float*)d_in[19];
  p.ag_b2    = (const float*)d_in[20];
  p.emb_is   = (const float*)d_in[21];
  p.emb_turn = (const float*)d_in[22];
  p.emb_tc   = (const float*)d_in[23];
  p.Wq       = (const float*)d_in[24];
  p.Wk       = (const float*)d_in[25];
  p.Wv       = (const float*)d_in[26];
  p.Wo       = (const float*)d_in[27];
  p.W_self   = (const float*)d_in[28];
  p.W_ih     = (const float*)d_in[29];
  p.W_hh     = (const float*)d_in[30];
  p.b_self   = (const float*)d_in[31];
  p.b_ih     = (const float*)d_in[32];
  p.b_hh     = (const float*)d_in[33];
  p.n1_g     = (const float*)d_in[34];
  p.n1_b     = (const float*)d_in[35];
  p.n2_g     = (const float*)d_in[36];
  p.n2_b     = (const float*)d_in[37];
  p.mlp_W1   = (const float*)d_in[38];
  p.mlp_b1   = (const float*)d_in[39];
  p.mlp_W2   = (const float*)d_in[40];
  p.mlp_b2   = (const float*)d_in[41];
  p.mlp_W3   = (const float*)d_in[42];
  p.mlp_b3   = (const float*)d_in[43];
  p.nbr_idx  = (const int*)d_in[44];
  p.is_inter = (const int*)d_in[45];
  p.turn_dir = (const int*)d_in[46];
  p.traffic  = (const int*)d_in[47];
  p.out      = (float*)d_out;

  // LDS: 7*4096 weights + 2176 small params + 4*BA*LS + 2*KNB*LS + 3*BA*LS
  constexpr size_t SMEM_FLOATS =
      7 * 4096 + 2176 + 4 * BA * LS + 2 * KNB * LS + 3 * BA * LS;
  dim3 grid(N_ACTOR_C / BA), block(NT);
  hipLaunchKernelGGL(alenc_kernel, grid, block, SMEM_FLOATS * sizeof(float),
                     stream, p);
}